// GATEncoder_54726473286270
// MI455X (gfx1250) — compile-verified
//
#include <hip/hip_runtime.h>
#include <hip/hip_bf16.h>
#include <math.h>

// Problem constants (match reference)
#define GB 4096      // batch (graphs)
#define GN 12        // nodes per graph
#define HID 64
#define NEG_SLOPE 0.2f

typedef __attribute__((ext_vector_type(2))) float    v2f;
typedef __attribute__((ext_vector_type(8))) float    v8f;
typedef __attribute__((ext_vector_type(4))) uint32_t v4u;
typedef __attribute__((ext_vector_type(8))) uint32_t v8u;

// LDS byte offset of a __shared__ object (generic -> addrspace(3) -> u32)
__device__ __forceinline__ uint32_t lds_offset(const void* p) {
    return (uint32_t)(uintptr_t)(__attribute__((address_space(3))) const char*)p;
}

// ---------------------------------------------------------------------------
// fp32 WMMA GEMM: C[M,N] = A[M,K] @ B[K,N] (+ bias[N] if non-null)
//  * One wave32 computes one 16x16 C tile via V_WMMA_F32_16X16X4_F32.
//  * 4 waves / block share one 16xK A tile, DMA'd into LDS by the Tensor Data
//    Mover (tensor_load_to_lds, TENSORcnt), with 2-DWORD padding every
//    64 DWORDs to stagger LDS banks (row stride K + 2*(K/64) floats, even).
//  * Requires M%16==0 and N == gridDim.y*64 (all launches below satisfy it),
//    so EXEC is all-ones at every WMMA and every wave reaches the barrier.
//
// f32 WMMA fragment layout (ISA 7.12.2):
//   A (16x4):  lane l -> M = l&15 ; VGPR j holds K = 2*(l>=16) + j
//   B (4x16):  lane l -> N = l&15 ; VGPR j holds K = 2*(l>=16) + j
//   C/D 16x16: VGPR r -> M = r + 8*(l>=16), N = l&15
// ---------------------------------------------------------------------------
template <int K>
__global__ void wmma_gemm_f32(const float* __restrict__ A,
                              const float* __restrict__ B,
                              const float* __restrict__ bias,
                              float* __restrict__ C,
                              int N) {
    constexpr int KP = K + 2 * (K / 64);        // padded LDS row stride (even)
    __shared__ float sh_a[16 * KP];

    const int wave   = threadIdx.x >> 5;        // 0..3
    const int lane   = threadIdx.x & 31;
    const int lane15 = lane & 15;
    const int khalf  = (lane >> 4) << 1;        // 0 or 2
    const int m0     = blockIdx.x << 4;
    const int n0     = ((blockIdx.y << 2) + wave) << 4;

    // ---- TDM: DMA the 16xK fp32 A tile into LDS (wave 0 only) -------------
    if (wave == 0) {
        const uint64_t ga = (uint64_t)(uintptr_t)(A + (size_t)m0 * K);
        v4u g0, g2, g3;
        v8u g1;
        // Group 0: count=1 | lds_addr | global_addr[56:0] | type=2
        g0[0] = 1u;
        g0[1] = lds_offset(sh_a);
        g0[2] = (uint32_t)ga;
        g0[3] = ((uint32_t)(ga >> 32) & 0x01FFFFFFu) | (2u << 30);
        // Group 1: data_size=2(4B) | pad_enable | pad_interval=5(64dw) |
        //          pad_amount=1(2dw) ; tensor 16 rows x K cols, tile = same
        const uint32_t td0 = (uint32_t)K;       // tensor_dim0 (elements)
        const uint32_t td1 = 16u;               // tensor_dim1 (rows)
        g1[0] = (2u << 16) | (1u << 20) | (5u << 22) | (1u << 25);
        g1[1] = (td0 & 0xFFFFu) << 16;                          // dim0[15:0]
        g1[2] = ((td0 >> 16) & 0xFFFFu) | ((td1 & 0xFFFFu) << 16);
        g1[3] = ((td1 >> 16) & 0xFFFFu) | ((uint32_t)K << 16);  // tile_dim0=K
        g1[4] = 16u;                            // tile_dim1=16, tile_dim2=0
        g1[5] = (uint32_t)K;                    // tensor_dim0_stride[31:0]
        g1[6] = 0u;                             // stride[47:32], dim1_stride lo
        g1[7] = 0u;
        g2 = (v4u){0u, 0u, 0u, 0u};             // 2D tensor: groups 2/3 unused
        g3 = (v4u){0u, 0u, 0u, 0u};
        asm volatile("tensor_load_to_lds %0, %1, %2, %3"
                     :
                     : "s"(g0), "s"(g1), "s"(g2), "s"(g3)
                     : "memory");
        __builtin_amdgcn_s_wait_tensorcnt(0);
    }
    __syncthreads();

    // ---- WMMA main loop ----------------------------------------------------
    const float* bcol = B + (size_t)khalf * N + n0 + lane15;

    v8f acc = {};
    for (int k = 0; k < K; k += 4) {
        const int ai = lane15 * KP + k + khalf + 2 * (k >> 6); // padded index
        v2f a = *(const v2f*)(sh_a + ai);       // 8B-aligned (KP even)
        v2f b;
        b.x = bcol[(size_t)k * N];              // B[k+khalf  ][n]
        b.y = bcol[(size_t)(k + 1) * N];        // B[k+khalf+1][n]
        acc = __builtin_amdgcn_wmma_f32_16x16x4_f32(
                  false, a, false, b, (short)0, acc, false, false);
    }

    const int mD = (lane >> 4) << 3;            // +8 for lanes 16..31
    const float bv = bias ? bias[n0 + lane15] : 0.0f;
    #pragma unroll
    for (int r = 0; r < 8; ++r) {
        C[(size_t)(m0 + mD + r) * N + n0 + lane15] = acc[r] + bv;
    }
}

// ---------------------------------------------------------------------------
// GAT layer 1 attention: 4 heads, C=64 per head. One block per graph.
// ---------------------------------------------------------------------------
__global__ void gat_att4(const float* __restrict__ h,        // [B*N, 256]
                         const float* __restrict__ att_src,  // [4,64]
                         const float* __restrict__ att_dst,  // [4,64]
                         const float* __restrict__ bias,     // [256]
                         float* __restrict__ out) {          // [B*N, 256]
    __shared__ float sh[GN * 256];
    __shared__ float s_as[GN * 4];
    __shared__ float s_ad[GN * 4];
    __shared__ float s_alpha[4][GN][GN];        // [head][dst][src]

    const int g = blockIdx.x;
    const int t = threadIdx.x;                  // 128 threads
    const float* hg = h + (size_t)g * GN * 256;

    for (int i = t; i < GN * 256; i += 128) sh[i] = hg[i];
    __syncthreads();

    if (t < GN * 4) {                           // 48 (node, head) dot pairs
        const int n = t >> 2, hd = t & 3;
        float as = 0.0f, ad = 0.0f;
        for (int c = 0; c < 64; ++c) {
            const float v = sh[n * 256 + hd * 64 + c];
            as += v * att_src[hd * 64 + c];
            ad += v * att_dst[hd * 64 + c];
        }
        s_as[t] = as;
        s_ad[t] = ad;
    }
    __syncthreads();

    if (t < GN * 4) {                           // per-(dst,head) softmax
        const int d = t >> 2, hd = t & 3;
        float e[GN];
        float m = -1e30f;
        for (int s = 0; s < GN; ++s) {
            float x = s_as[s * 4 + hd] + s_ad[d * 4 + hd];
            x = (x > 0.0f) ? x : x * NEG_SLOPE;
            e[s] = x;
            m = fmaxf(m, x);
        }
        float sum = 0.0f;
        for (int s = 0; s < GN; ++s) { e[s] = expf(e[s] - m); sum += e[s]; }
        const float inv = 1.0f / sum;
        for (int s = 0; s < GN; ++s) s_alpha[hd][d][s] = e[s] * inv;
    }
    __syncthreads();

    for (int i = t; i < GN * 256; i += 128) {   // aggregate + bias + ELU
        const int d = i >> 8;
        const int col = i & 255;
        const int hd = col >> 6;
        float acc = 0.0f;
        for (int s = 0; s < GN; ++s)
            acc = fmaf(s_alpha[hd][d][s], sh[s * 256 + col], acc);
        acc += bias[col];
        out[(size_t)g * GN * 256 + i] = (acc > 0.0f) ? acc : (expf(acc) - 1.0f);
    }
}

// ---------------------------------------------------------------------------
// GAT layer 2 attention: 1 head, C=64. One block per graph.
// ---------------------------------------------------------------------------
__global__ void gat_att1(const float* __restrict__ h,        // [B*N, 64]
                         const float* __restrict__ att_src,  // [64]
                         const float* __restrict__ att_dst,  // [64]
                         const float* __restrict__ bias,     // [64]
                         float* __restrict__ out) {          // [B*N, 64]
    __shared__ float sh[GN * 64];
    __shared__ float s_as[GN];
    __shared__ float s_ad[GN];
    __shared__ float s_alpha[GN][GN];

    const int g = blockIdx.x;
    const int t = threadIdx.x;                  // 128 threads
    const float* hg = h + (size_t)g * GN * 64;

    for (int i = t; i < GN * 64; i += 128) sh[i] = hg[i];
    __syncthreads();

    if (t < GN) {
        float as = 0.0f, ad = 0.0f;
        for (int c = 0; c < 64; ++c) {
            const float v = sh[t * 64 + c];
            as += v * att_src[c];
            ad += v * att_dst[c];
        }
        s_as[t] = as;
        s_ad[t] = ad;
    }
    __syncthreads();

    if (t < GN) {
        const int d = t;
        float e[GN];
        float m = -1e30f;
        for (int s = 0; s < GN; ++s) {
            float x = s_as[s] + s_ad[d];
            x = (x > 0.0f) ? x : x * NEG_SLOPE;
            e[s] = x;
            m = fmaxf(m, x);
        }
        float sum = 0.0f;
        for (int s = 0; s < GN; ++s) { e[s] = expf(e[s] - m); sum += e[s]; }
        const float inv = 1.0f / sum;
        for (int s = 0; s < GN; ++s) s_alpha[d][s] = e[s] * inv;
    }
    __syncthreads();

    for (int i = t; i < GN * 64; i += 128) {
        const int d = i >> 6;
        const int c = i & 63;
        float acc = 0.0f;
        for (int s = 0; s < GN; ++s)
            acc = fmaf(s_alpha[d][s], sh[s * 64 + c], acc);
        acc += bias[c];
        out[(size_t)g * GN * 64 + i] = (acc > 0.0f) ? acc : (expf(acc) - 1.0f);
    }
}

// ---------------------------------------------------------------------------
// Launch: GEMM1 -> att(4h) -> GEMM2 -> att(1h) -> proj GEMM (+bias)
// Inputs: 0 x_batch, 1 adj(ignored, dense), 2 W1, 3 att_src1, 4 att_dst1,
//         5 b1, 6 W2, 7 att_src2, 8 att_dst2, 9 b2, 10 Wp, 11 bp
// ---------------------------------------------------------------------------
extern "C" void kernel_launch(void* const* d_in, const int* in_sizes, int n_in,
                              void* d_out, int out_size, void* d_ws, size_t ws_size,
                              hipStream_t stream) {
    (void)in_sizes; (void)n_in; (void)out_size; (void)ws_size;

    const float* x   = (const float*)d_in[0];
    const float* W1  = (const float*)d_in[2];
    const float* as1 = (const float*)d_in[3];
    const float* ad1 = (const float*)d_in[4];
    const float* b1  = (const float*)d_in[5];
    const float* W2  = (const float*)d_in[6];
    const float* as2 = (const float*)d_in[7];
    const float* ad2 = (const float*)d_in[8];
    const float* b2  = (const float*)d_in[9];
    const float* Wp  = (const float*)d_in[10];
    const float* bp  = (const float*)d_in[11];
    float* out = (float*)d_out;

    const int NT = GB * GN;                       // 49152 total nodes
    float* buf0 = (float*)d_ws;                   // up to [NT, 256]
    float* buf1 = buf0 + (size_t)NT * 256;        // up to [NT, 256]

    const dim3 blk(128);

    // 1) h1 = X @ W1 : [49152,64] @ [64,256]
    wmma_gemm_f32<64><<<dim3(NT / 16, 4), blk, 0, stream>>>(
        x, W1, nullptr, buf0, 256);

    // 2) x1 = elu(att4(h1) + b1) : [49152,256]
    gat_att4<<<GB, 128, 0, stream>>>(buf0, as1, ad1, b1, buf1);

    // 3) h2 = x1 @ W2 : [49152,256] @ [256,64]
    wmma_gemm_f32<256><<<dim3(NT / 16, 1), blk, 0, stream>>>(
        buf1, W2, nullptr, buf0, 64);

    // 4) x2 = elu(att1(h2) + b2) : [49152,64]
    gat_att1<<<GB, 128, 0, stream>>>(buf0, as2, ad2, b2, buf1);

    // 5) out = x2.reshape(4096,768) @ Wp + bp : [4096,128]
    wmma_gemm_f32<768><<<dim3(GB / 16, 2), blk, 0, stream>>>(
        buf1, Wp, bp, out, 128);
}